// GNNHist_50268297232463
// MI455X (gfx1250) — compile-verified
//
#include <hip/hip_runtime.h>
#include <hip/hip_bf16.h>
#include <math.h>

// Problem constants (from reference)
#define BB   256          // batch
#define U1N  160          // candidate nodes
#define VN   16           // decode steps
#define DN   128          // embedding dim
#define GN   (U1N + VN)   // 176
#define HN   200          // hidden
#define HP   224          // hidden padded to 14 * 16 (WMMA N tiles)
#define RN   (BB * U1N)   // 40960 MLP rows

typedef float v2f __attribute__((ext_vector_type(2)));
typedef float v8f __attribute__((ext_vector_type(8)));

__device__ __forceinline__ v8f wmma_f32(v2f a, v2f b, v8f c) {
  // D = A(16x4,f32) * B(4x16,f32) + C(16x16,f32)
  return __builtin_amdgcn_wmma_f32_16x16x4_f32(
      /*neg_a=*/false, a, /*neg_b=*/false, b,
      /*c_mod=*/(short)0, c, /*reuse_a=*/false, /*reuse_b=*/false);
}

// ---------------- workspace layout (float units) ----------------
#define WS_MEANU 0                          // B*D           = 32768
#define WS_WCAT  (WS_MEANU + BB * DN)       // 256*224       = 57344
#define WS_W2P   (WS_WCAT + 2 * DN * HP)    // 200*224       = 44800
#define WS_CV    (WS_W2P + HN * HP)         // B*V*D         = 524288
#define WS_S     (WS_CV + BB * VN * DN)     // B*D
#define WS_SIZE  (WS_S + BB * DN)           // B
#define WS_G     (WS_SIZE + BB)             // B*H
#define WS_PI0   (WS_G + BB * HN)           // R
#define WS_PI1   (WS_PI0 + RN)              // R
#define WS_SEL   (WS_PI1 + RN)              // B*V ints
#define WS_HSTAT (WS_SEL + BB * VN)         // R*H  (~32.8 MB)

// ---------------- init kernels (run every launch: determinism) ----------------
__global__ void k_init_state(float* __restrict__ S, float* __restrict__ sizev,
                             int* __restrict__ selh) {
  int i = blockIdx.x * blockDim.x + threadIdx.x;
  if (i < BB * DN) S[i] = 0.f;
  if (i < BB)      sizev[i] = 0.f;
  if (i < BB * VN) selh[i] = 0;
}

// Pack Wcat (u_emb rows ++ mean_u rows of W1) and W2, N-padded to 224 with zeros.
__global__ void k_pack(const float* __restrict__ W1, const float* __restrict__ W2,
                       float* __restrict__ Wcat, float* __restrict__ W2p) {
  int i = blockIdx.x * blockDim.x + threadIdx.x;
  if (i < 2 * DN * HP) {
    int k = i / HP, j = i % HP;
    float v = 0.f;
    if (j < HN)
      v = (k < DN) ? W1[(5 + DN + k) * HN + j]             // u_emb rows 133..260
                   : W1[(5 + 3 * DN + (k - DN)) * HN + j];  // mean_u rows 389..516
    Wcat[i] = v;
  }
  if (i < HN * HP) {
    int k = i / HP, j = i % HP;
    W2p[i] = (j < HN) ? W2[k * HN + j] : 0.f;
  }
}

__global__ void k_mean(const float* __restrict__ emb, float* __restrict__ meanu) {
  int b = blockIdx.x, d = threadIdx.x;  // 128 threads
  float s = 0.f;
  for (int u = 0; u < U1N; ++u) s += emb[((size_t)b * GN + u) * DN + d];
  meanu[b * DN + d] = s * (1.f / (float)U1N);
}

// cv[b,j,:] = v_emb[b,j,:] @ Wsc[D:2D,:] + bsc  (step-invariant half of ctx)
__global__ void k_cv(const float* __restrict__ emb, const float* __restrict__ Wsc,
                     const float* __restrict__ bsc, float* __restrict__ cv) {
  int bj = blockIdx.x;            // b*16 + j
  int b = bj >> 4, j = bj & 15;
  int d = threadIdx.x;            // 128 threads
  float acc = bsc[d];
  const float* er = emb + ((size_t)b * GN + U1N + j) * DN;
  for (int d2 = 0; d2 < DN; ++d2) acc += er[d2] * Wsc[(DN + d2) * DN + d];
  cv[(size_t)bj * DN + d] = acc;
}

// ---------------- WMMA GEMM 1: h_static = [u_emb | mean_u] @ Wcat + b1 + fni ----------------
#define KS1 32
__global__ __launch_bounds__(256) void k_gemm_static(
    const float* __restrict__ emb, const float* __restrict__ W1,
    const float* __restrict__ b1, const float* __restrict__ meanu,
    const float* __restrict__ Wcat, float* __restrict__ hstat) {
  // K-pair-major weight slice: B frag = contiguous float2 -> ds_load_b64
  __shared__ float sW[KS1 / 2][HP][2];   // 28.0 KB
  __shared__ float sX[64][KS1 + 2];      //  8.5 KB (padded stride: bank spread)
  const int tid = threadIdx.x;
  const int lane = tid & 31;
  const int wid = tid >> 5;       // 0..7
  const int rs = wid & 3;         // row subtile within 64-row macro tile
  const int ch = wid >> 2;        // col half (0/1) -> 7 tiles each
  const int half = lane >> 4;     // 0/1
  const int m = lane & 15;
  const int rowBase = blockIdx.x * 64;

  v8f acc[7];
  for (int tt = 0; tt < 7; ++tt)
    for (int i = 0; i < 8; ++i) acc[tt][i] = 0.f;

  for (int ks = 0; ks < 2 * DN; ks += KS1) {   // K = 256, 8 slices
    if (ks + KS1 < 2 * DN)
      __builtin_prefetch(&Wcat[(size_t)(ks + KS1) * HP], 0, 0);
    for (int idx = tid; idx < KS1 * HP; idx += 256) {
      int k = idx / HP, j = idx % HP;
      sW[k >> 1][j][k & 1] = Wcat[(size_t)(ks + k) * HP + j];
    }
    for (int idx = tid; idx < 64 * KS1; idx += 256) {
      int r = idx / KS1, k = idx % KS1;
      int row = rowBase + r;
      int b = row / U1N, u = row - b * U1N;
      int kg = ks + k;
      sX[r][k] = (kg < DN) ? emb[((size_t)b * GN + u) * DN + kg]
                           : meanu[b * DN + (kg - DN)];
    }
    __syncthreads();
    for (int kk = 0; kk < KS1; kk += 4) {
      v2f a = *(const v2f*)&sX[rs * 16 + m][kk + 2 * half];
#pragma unroll
      for (int tt = 0; tt < 7; ++tt) {
        int col = ch * 112 + tt * 16 + m;
        v2f bv = *(const v2f*)&sW[(kk >> 1) + half][col][0];
        acc[tt] = wmma_f32(a, bv, acc[tt]);
      }
    }
    __syncthreads();
  }
  // epilogue: + b1 + fni(u==0)*W1[3,:]
  for (int tt = 0; tt < 7; ++tt) {
    int col = ch * 112 + tt * 16 + m;
    if (col < HN) {
#pragma unroll
      for (int i = 0; i < 8; ++i) {
        int row = rowBase + rs * 16 + i + 8 * half;
        int u = row % U1N;
        float v = acc[tt][i] + b1[col];
        if (u == 0) v += W1[3 * HN + col];
        hstat[(size_t)row * HN + col] = v;
      }
    }
  }
}

// ---------------- per-step prep: g[b,:] = tf*W1r1 + sf*W1r2 + incoming@W1 + sc@W1 ----------------
__global__ __launch_bounds__(256) void k_prep(
    const float* __restrict__ emb, const float* __restrict__ W1,
    const float* __restrict__ S, const float* __restrict__ sizev,
    const float* __restrict__ ictx, float* __restrict__ g, int t) {
  __shared__ float inc_s[DN], sc_s[DN];
  int b = blockIdx.x, tid = threadIdx.x;
  if (tid < DN) {
    inc_s[tid] = emb[((size_t)b * GN + (U1N + t)) * DN + tid];
    sc_s[tid] = (t == 0) ? ictx[tid] : S[b * DN + tid] / (float)t;
  }
  __syncthreads();
  if (tid < HN) {
    float tf = (float)t / (float)VN;
    float sf = sizev[b] / (float)(U1N - 1);
    float acc = tf * W1[1 * HN + tid] + sf * W1[2 * HN + tid];
    for (int d2 = 0; d2 < DN; ++d2) {
      acc += inc_s[d2] * W1[(5 + d2) * HN + tid];             // incoming rows
      acc += sc_s[d2]  * W1[(5 + 2 * DN + d2) * HN + tid];    // sc rows
    }
    g[b * HN + tid] = acc;
  }
}

// ---------------- per-step WMMA GEMM: piPart = relu(relu(h1)@W2 + b2) @ W3 ----------------
#define KS2 40
__global__ __launch_bounds__(256) void k_gemm_l2(
    const float* __restrict__ hstat, const float* __restrict__ g,
    const float* __restrict__ adj, const int* __restrict__ mask,
    const float* __restrict__ W1, const float* __restrict__ W2p,
    const float* __restrict__ b2, const float* __restrict__ W3,
    float* __restrict__ pi0, float* __restrict__ pi1, int t) {
  __shared__ float sW[KS2 / 2][HP][2];   // 35.0 KB
  __shared__ float sX[64][KS2 + 2];      // 10.5 KB
  const int tid = threadIdx.x;
  const int lane = tid & 31;
  const int wid = tid >> 5;
  const int rs = wid & 3;
  const int ch = wid >> 2;
  const int half = lane >> 4;
  const int m = lane & 15;
  const int rowBase = blockIdx.x * 64;

  v8f acc[7];
  for (int tt = 0; tt < 7; ++tt)
    for (int i = 0; i < 8; ++i) acc[tt][i] = 0.f;

  for (int ks = 0; ks < HN; ks += KS2) {  // K = 200, 5 slices, K%4==0: no padding
    if (ks + KS2 < HN)
      __builtin_prefetch(&W2p[(size_t)(ks + KS2) * HP], 0, 0);
    for (int idx = tid; idx < KS2 * HP; idx += 256) {
      int k = idx / HP, j = idx % HP;
      sW[k >> 1][j][k & 1] = W2p[(size_t)(ks + k) * HP + j];
    }
    // h1 materialized on the fly: relu(h_static + g + w*W1r0 + m*W1r4)
    for (int idx = tid; idx < 64 * KS2; idx += 256) {
      int r = idx / KS2, k = idx % KS2;
      int row = rowBase + r;
      int b = row / U1N, u = row - b * U1N;
      int kg = ks + k;
      size_t mi = ((size_t)b * VN + t) * U1N + u;
      float wv = adj[mi];
      float mv = (float)mask[mi];
      float v = hstat[(size_t)row * HN + kg] + g[b * HN + kg] +
                wv * W1[kg] + mv * W1[4 * HN + kg];
      sX[r][k] = fmaxf(v, 0.f);
    }
    __syncthreads();
    for (int kk = 0; kk < KS2; kk += 4) {
      v2f a = *(const v2f*)&sX[rs * 16 + m][kk + 2 * half];
#pragma unroll
      for (int tt = 0; tt < 7; ++tt) {
        int col = ch * 112 + tt * 16 + m;
        v2f bv = *(const v2f*)&sW[(kk >> 1) + half][col][0];
        acc[tt] = wmma_f32(a, bv, acc[tt]);
      }
    }
    __syncthreads();
  }
  // fused layer-3: local[i] = sum over this wave's cols of relu(acc+b2)*W3
  float local[8];
#pragma unroll
  for (int i = 0; i < 8; ++i) local[i] = 0.f;
  for (int tt = 0; tt < 7; ++tt) {
    int col = ch * 112 + tt * 16 + m;
    float w3 = (col < HN) ? W3[col] : 0.f;
    float bb = (col < HN) ? b2[col] : 0.f;
#pragma unroll
    for (int i = 0; i < 8; ++i)
      local[i] += fmaxf(acc[tt][i] + bb, 0.f) * w3;
  }
  // reduce across the 16 lanes sharing each output row; each col-half wave
  // writes its own partial buffer (no atomics, no init kernel needed)
  float* piPart = (ch == 0) ? pi0 : pi1;
#pragma unroll
  for (int i = 0; i < 8; ++i) {
    float v = local[i];
    for (int off = 1; off < 16; off <<= 1) v += __shfl_xor(v, off, 16);
    if (m == 0) {
      int row = rowBase + rs * 16 + i + 8 * half;
      piPart[row] = v;
    }
  }
}

// ---------------- per-step select: masked log_softmax, argmax, state update ----------------
__global__ __launch_bounds__(256) void k_select(
    const float* __restrict__ pi0, const float* __restrict__ pi1,
    const float* __restrict__ b3, const int* __restrict__ mask,
    const float* __restrict__ adj, const float* __restrict__ emb,
    const float* __restrict__ Wsc, const float* __restrict__ cv,
    float* __restrict__ S, float* __restrict__ sizev, int* __restrict__ selh,
    float* __restrict__ out, int t) {
  __shared__ float pv[U1N];
  __shared__ float rmax[256];
  __shared__ int   ridx[256];
  __shared__ float rsum[256];
  __shared__ float smx, slse;
  __shared__ int   ssel;
  int b = blockIdx.x, tid = threadIdx.x;
  float p = -3.402823466e38f;
  if (tid < U1N) {
    int ri = b * U1N + tid;
    float raw = pi0[ri] + pi1[ri] + b3[0];
    int mm = mask[((size_t)b * VN + t) * U1N + tid];
    p = (mm > 0) ? -1000000.0f : raw;
    pv[tid] = p;
  }
  rmax[tid] = p;
  ridx[tid] = (tid < U1N) ? tid : 0x7fffffff;
  __syncthreads();
  for (int s = 128; s > 0; s >>= 1) {
    if (tid < s) {
      float av = rmax[tid], bv = rmax[tid + s];
      int ai = ridx[tid], bi = ridx[tid + s];
      if (bv > av || (bv == av && bi < ai)) { rmax[tid] = bv; ridx[tid] = bi; }
    }
    __syncthreads();
  }
  if (tid == 0) { smx = rmax[0]; ssel = ridx[0]; }
  __syncthreads();
  float mx = smx;
  rsum[tid] = (tid < U1N) ? expf(pv[tid] - mx) : 0.f;
  __syncthreads();
  for (int s = 128; s > 0; s >>= 1) {
    if (tid < s) rsum[tid] += rsum[tid + s];
    __syncthreads();
  }
  if (tid == 0) slse = logf(rsum[0]);
  __syncthreads();
  if (tid < U1N)
    out[((size_t)b * VN + t) * U1N + tid] = pv[tid] - mx - slse;
  int sel = ssel;
  if (tid == 0) {
    float ns = sizev[b] + adj[((size_t)b * VN + t) * U1N + sel];
    sizev[b] = ns;
    selh[b * VN + t] = sel;
    out[(size_t)BB * VN * U1N + b * VN + t] = (float)sel;          // sequences
    out[(size_t)BB * VN * U1N + BB * VN + b] = ns;                 // size (final wins)
  }
  // ctx running sum: S += emb[b,sel] @ Wsc[0:D] + cv[b,t]
  if (tid < DN) {
    float acc = cv[((size_t)b * VN + t) * DN + tid];
    const float* er = emb + ((size_t)b * GN + sel) * DN;
    for (int d2 = 0; d2 < DN; ++d2) acc += er[d2] * Wsc[d2 * DN + tid];
    S[b * DN + tid] += acc;
  }
}

// ---------------- host launcher ----------------
extern "C" void kernel_launch(void* const* d_in, const int* in_sizes, int n_in,
                              void* d_out, int out_size, void* d_ws, size_t ws_size,
                              hipStream_t stream) {
  const float* emb  = (const float*)d_in[0];
  const float* adj  = (const float*)d_in[1];
  const int*   mask = (const int*)  d_in[2];
  const float* W1   = (const float*)d_in[3];
  const float* b1   = (const float*)d_in[4];
  const float* W2   = (const float*)d_in[5];
  const float* b2   = (const float*)d_in[6];
  const float* W3   = (const float*)d_in[7];
  const float* b3   = (const float*)d_in[8];
  const float* Wsc  = (const float*)d_in[9];
  const float* bsc  = (const float*)d_in[10];
  const float* ictx = (const float*)d_in[11];
  float* out = (float*)d_out;
  float* ws  = (float*)d_ws;

  float* meanu = ws + WS_MEANU;
  float* Wcat  = ws + WS_WCAT;
  float* W2p   = ws + WS_W2P;
  float* cv    = ws + WS_CV;
  float* S     = ws + WS_S;
  float* sizev = ws + WS_SIZE;
  float* g     = ws + WS_G;
  float* pi0   = ws + WS_PI0;
  float* pi1   = ws + WS_PI1;
  int*   selh  = (int*)(ws + WS_SEL);
  float* hstat = ws + WS_HSTAT;

  k_init_state<<<(BB * DN + 255) / 256, 256, 0, stream>>>(S, sizev, selh);
  k_pack<<<(2 * DN * HP + 255) / 256, 256, 0, stream>>>(W1, W2, Wcat, W2p);
  k_mean<<<BB, DN, 0, stream>>>(emb, meanu);
  k_cv<<<BB * VN, DN, 0, stream>>>(emb, Wsc, bsc, cv);
  k_gemm_static<<<RN / 64, 256, 0, stream>>>(emb, W1, b1, meanu, Wcat, hstat);

  for (int t = 0; t < VN; ++t) {
    k_prep<<<BB, 256, 0, stream>>>(emb, W1, S, sizev, ictx, g, t);
    k_gemm_l2<<<RN / 64, 256, 0, stream>>>(hstat, g, adj, mask, W1, W2p, b2, W3,
                                           pi0, pi1, t);
    k_select<<<BB, 256, 0, stream>>>(pi0, pi1, b3, mask, adj, emb, Wsc, cv, S,
                                     sizev, selh, out, t);
  }
}